// AELoss_69655779606927
// MI455X (gfx1250) — compile-verified
//
#include <hip/hip_runtime.h>
#include <math.h>

// Problem constants (from reference setup_inputs)
#define DIM_B 2048
#define DIM_C 3
#define DIM_T 300
#define DIM_J 25
#define TILE   (DIM_T * DIM_J)   // 7500 floats per (b,c) tile
#define NVEC   (TILE / 4)        // 1875 float4 vectors per tile
#define NTILES (DIM_B * DIM_C)   // 6144 tiles
#define NTHR   256               // 8 wave32 per block

#define AS1 __attribute__((address_space(1)))
#define AS3 __attribute__((address_space(3)))

typedef int v4i __attribute__((ext_vector_type(4)));

// ---- CDNA5 async global->LDS copy (128-bit), with fallback -----------------
__device__ __forceinline__ void async_copy16(const float* g, float* l) {
#if __has_builtin(__builtin_amdgcn_global_load_async_to_lds_b128)
    // (global int4* src, lds int4* dst, imm offset, imm cpol)
    __builtin_amdgcn_global_load_async_to_lds_b128(
        (AS1 v4i*)g, (AS3 v4i*)l, 0, 0);
#else
    *(float4*)l = *(const float4*)g;   // fallback: load + ds_store
#endif
}

__device__ __forceinline__ void wait_async_zero() {
#if __has_builtin(__builtin_amdgcn_s_wait_asynccnt)
    __builtin_amdgcn_s_wait_asynccnt(0);
#elif defined(__gfx1250__)
    asm volatile("s_wait_asynccnt 0" ::: "memory");
#endif
}

// ---- Kernel 1: one workgroup per (b,c) tile --------------------------------
__global__ __launch_bounds__(NTHR) void aeloss_tile_kernel(
        const float* __restrict__ x, const float* __restrict__ y,
        float* __restrict__ prec, float* __restrict__ psmooth) {
    __shared__ __align__(16) float lx[TILE];
    __shared__ __align__(16) float ly[TILE];
    __shared__ float part[4][8][32];   // S1x,S2x,S1y,S2y per wave per lane(j)
    __shared__ float recpart[8];

    const int tid  = threadIdx.x;
    const long long base = (long long)blockIdx.x * TILE;
    const float* gx = x + base;
    const float* gy = y + base;

    // Stage both 30KB tiles into LDS with async 128b copies (coalesced).
    for (int v = tid; v < NVEC; v += NTHR) {
        async_copy16(gx + 4 * v, &lx[4 * v]);
        async_copy16(gy + 4 * v, &ly[4 * v]);
    }
    wait_async_zero();
    __syncthreads();

    const int lane = tid & 31;   // j index (lanes 25..31 idle)
    const int wave = tid >> 5;   // t-stripe 0..7

    float s1x = 0.f, s2x = 0.f, s1y = 0.f, s2y = 0.f, rec = 0.f;
    if (lane < DIM_J) {
        for (int t = wave; t < DIM_T; t += 8) {
            float xv = lx[t * DIM_J + lane];
            float yv = ly[t * DIM_J + lane];
            float d  = xv - yv;
            rec += d * d;                       // reconstruction term
            s1x += xv; s2x += xv * xv;          // running sum / sum-of-squares
            s1y += yv; s2y += yv * yv;
        }
    }
    part[0][wave][lane] = s1x;
    part[1][wave][lane] = s2x;
    part[2][wave][lane] = s1y;
    part[3][wave][lane] = s2y;

    // wave32 reduce of the rec partial
    for (int o = 16; o > 0; o >>= 1) rec += __shfl_xor(rec, o, 32);
    if (lane == 0) recpart[wave] = rec;
    __syncthreads();

    if (wave == 0) {
        float S1x = 0.f, S2x = 0.f, S1y = 0.f, S2y = 0.f;
#pragma unroll
        for (int w = 0; w < 8; ++w) {
            S1x += part[0][w][lane];
            S2x += part[1][w][lane];
            S1y += part[2][w][lane];
            S2y += part[3][w][lane];
        }
        // sx[j] = sum_{t<=298} x[t,j] - sum_{t>=1} x[t,j]^2
        //       = (S1 - x[299,j]) - (S2 - x[0,j]^2); only j = 0..J-2 counted.
        float dabs = 0.f;
        if (lane < DIM_J - 1) {
            float xl = lx[(DIM_T - 1) * DIM_J + lane], xf = lx[lane];
            float yl = ly[(DIM_T - 1) * DIM_J + lane], yf = ly[lane];
            float sxj = (S1x - xl) - (S2x - xf * xf);
            float stj = (S1y - yl) - (S2y - yf * yf);
            dabs = fabsf(sxj - stj);
        }
        for (int o = 16; o > 0; o >>= 1) dabs += __shfl_xor(dabs, o, 32);

        if (lane == 0) {
            float rsum = 0.f;
#pragma unroll
            for (int w = 0; w < 8; ++w) rsum += recpart[w];
            prec[blockIdx.x]    = rsum;
            psmooth[blockIdx.x] = sqrtf(dabs) * (1.0f / (float)(DIM_J * DIM_T));
        }
    }
}

// ---- Kernel 2: deterministic final reduction over 6144 tile partials -------
__global__ __launch_bounds__(NTHR) void aeloss_final_kernel(
        const float* __restrict__ prec, const float* __restrict__ psmooth,
        float* __restrict__ out) {
    __shared__ float sr[8], ss[8];
    float r = 0.f, s = 0.f;
    for (int i = threadIdx.x; i < NTILES; i += NTHR) {
        r += prec[i];
        s += psmooth[i];
    }
    for (int o = 16; o > 0; o >>= 1) {
        r += __shfl_xor(r, o, 32);
        s += __shfl_xor(s, o, 32);
    }
    const int lane = threadIdx.x & 31, wave = threadIdx.x >> 5;
    if (lane == 0) { sr[wave] = r; ss[wave] = s; }
    __syncthreads();
    if (threadIdx.x == 0) {
        float R = 0.f, S = 0.f;
#pragma unroll
        for (int w = 0; w < 8; ++w) { R += sr[w]; S += ss[w]; }
        const float loss_rec    = R / (float)((long long)DIM_B * DIM_C * DIM_T * DIM_J);
        const float loss_smooth = S / (float)NTILES;
        out[0] = 2.0f * loss_rec + 3.0f * loss_smooth;
    }
}

extern "C" void kernel_launch(void* const* d_in, const int* in_sizes, int n_in,
                              void* d_out, int out_size, void* d_ws, size_t ws_size,
                              hipStream_t stream) {
    const float* x = (const float*)d_in[0];   // inputs  [2048,3,300,25] f32
    const float* y = (const float*)d_in[1];   // targets [2048,3,300,25] f32
    float* prec    = (float*)d_ws;            // 6144 rec partials
    float* psmooth = prec + NTILES;           // 6144 smooth partials

    aeloss_tile_kernel<<<NTILES, NTHR, 0, stream>>>(x, y, prec, psmooth);
    aeloss_final_kernel<<<1, NTHR, 0, stream>>>(prec, psmooth, (float*)d_out);
}